// Net_17394617549529
// MI455X (gfx1250) — compile-verified
//
#include <hip/hip_runtime.h>

typedef __attribute__((ext_vector_type(2))) float v2f;
typedef __attribute__((ext_vector_type(8))) float v8f;

#define H_IN 512
#define W_IN 512
#define PH   255
#define PW   255
#define NCB  16      // ceil(255/16) column blocks of 16 pooled columns
#define NCH  5

// One wave (32 lanes) computes 16 pooled columns x 5 channels for one
// (batch, pooled_row). Conv(3x3,VALID) + bias + 2x2 maxpool + ReLU fused.
// Four WMMA accumulator chains = the four pool quadrants; pooling is an
// elementwise max of D matrices (no cross-lane ops).
__global__ __launch_bounds__(256) void conv_pool_relu_wmma(
    const float* __restrict__ x,     // (64,1,512,512)
    const float* __restrict__ wk,    // (5,1,3,3)
    const float* __restrict__ bias,  // (5,)
    float* __restrict__ out)         // (64,5,255,255)
{
    const int lane = threadIdx.x & 31;
    const int wave = threadIdx.x >> 5;
    const int task = blockIdx.x * 8 + wave;

    const int b   = task / (PH * NCB);
    const int rem = task % (PH * NCB);
    const int ph  = rem / NCB;
    const int pw0 = (rem % NCB) * 16;

    const int  cw = lane & 15;      // pooled column within block / A-matrix row
    const bool up = lane >= 16;     // upper half-wave holds K=2,3 of each chunk

    // ---- A matrix: 16x12 weights (rows 5..15 zero, K col 4j+3 zero) ----
    // chunk j = kernel row; lower lanes: (w[c][j][0], w[c][j][1]);
    // upper lanes: (w[c][j][2], 0)
    v2f a0 = {0.f, 0.f}, a1 = {0.f, 0.f}, a2 = {0.f, 0.f};
    if (cw < NCH) {
        const float* wr = wk + cw * 9;
        if (up) {
            a0 = {wr[2], 0.f};
            a1 = {wr[5], 0.f};
            a2 = {wr[8], 0.f};
        } else {
            a0 = {wr[0], wr[1]};
            a1 = {wr[3], wr[4]};
            a2 = {wr[6], wr[7]};
        }
    }

    // ---- load 4 input floats from each of 4 rows (2 aligned b64 loads) ----
    const int pw = pw0 + cw;
    int cs  = 2 * pw + (up ? 2 : 0);              // always even -> 8B aligned
    int csc = cs > (W_IN - 4) ? (W_IN - 4) : cs;  // clamp, never crosses row end
    const int delta = cs - csc;                    // 0, 2, or 4 (4 only if masked)

    const float* rowp = x + ((size_t)b * H_IN + 2 * (size_t)ph) * W_IN + csc;

    float rv[4][4];
#pragma unroll
    for (int i = 0; i < 4; ++i) {
        v2f lo = *(const v2f*)(rowp + (size_t)i * W_IN);
        v2f hi = *(const v2f*)(rowp + (size_t)i * W_IN + 2);
        rv[i][0] = lo.x; rv[i][1] = lo.y; rv[i][2] = hi.x; rv[i][3] = hi.y;
        if (delta >= 2) {   // recover cols 510/511 for valid edge lanes
            rv[i][0] = rv[i][2];
            rv[i][1] = rv[i][3];
        }
    }

    // Prefetch the rows the next pooled-row tasks will need.
    if (2 * ph + 7 < H_IN) {
        __builtin_prefetch(rowp + 4 * (size_t)W_IN, 0, 1);
    }

    // ---- 4 quadrant accumulators; 3 chained K=4 WMMAs each ----
    v8f d00 = {}, d01 = {}, d10 = {}, d11 = {};

#pragma unroll
    for (int j = 0; j < 3; ++j) {
        const v2f aj = (j == 0) ? a0 : (j == 1) ? a1 : a2;
        v2f b00 = {rv[j][0],     rv[j][1]};      // dy=0, dx=0
        v2f b01 = {rv[j][1],     rv[j][2]};      // dy=0, dx=1
        v2f b10 = {rv[j + 1][0], rv[j + 1][1]};  // dy=1, dx=0
        v2f b11 = {rv[j + 1][1], rv[j + 1][2]};  // dy=1, dx=1
        d00 = __builtin_amdgcn_wmma_f32_16x16x4_f32(false, aj, false, b00,
                                                    (short)0, d00, false, false);
        d01 = __builtin_amdgcn_wmma_f32_16x16x4_f32(false, aj, false, b01,
                                                    (short)0, d01, false, false);
        d10 = __builtin_amdgcn_wmma_f32_16x16x4_f32(false, aj, false, b10,
                                                    (short)0, d10, false, false);
        d11 = __builtin_amdgcn_wmma_f32_16x16x4_f32(false, aj, false, b11,
                                                    (short)0, d11, false, false);
    }

    // ---- maxpool (elementwise over quadrants) + bias + ReLU + store ----
    // Lanes 0..15, D VGPR v (v<5) = channel v at pooled column pw.
    if (!up && pw < PW) {
        float* o = out + (((size_t)b * NCH) * PH + ph) * PW + pw;
#pragma unroll
        for (int c = 0; c < NCH; ++c) {
            float m = fmaxf(fmaxf(d00[c], d01[c]), fmaxf(d10[c], d11[c]));
            m = fmaxf(m + bias[c], 0.f);
            o[(size_t)c * (PH * PW)] = m;
        }
    }
}

extern "C" void kernel_launch(void* const* d_in, const int* in_sizes, int n_in,
                              void* d_out, int out_size, void* d_ws, size_t ws_size,
                              hipStream_t stream) {
    const float* x    = (const float*)d_in[0];
    const float* wk   = (const float*)d_in[1];
    const float* bias = (const float*)d_in[2];
    float* out        = (float*)d_out;

    const int tasks  = 64 * PH * NCB;   // one wave per task
    const int blocks = tasks / 8;       // 8 waves per 256-thread block = 32640
    conv_pool_relu_wmma<<<blocks, 256, 0, stream>>>(x, wk, bias, out);
}